// Conv_53798760350153
// MI455X (gfx1250) — compile-verified
//
#include <hip/hip_runtime.h>
#include <cstdint>

// ---------------------------------------------------------------------------
// Depthwise 5x5 conv (NCHW, pad=2) + bias, fp32.  HBM-bound: ~822MB traffic,
// 5.14 GFLOP -> ~35us floor at 23.3 TB/s.  CDNA5 path: async global->LDS
// staging (ASYNCcnt, b64-wide) of the halo tile, then pure-FMA stencil
// compute out of LDS with conflict-free banks and coalesced 128B stores.
// ---------------------------------------------------------------------------

namespace {
constexpr int Bn = 16, Cc = 128, Hh = 224, Ww = 224;
constexpr int TS = 32;              // output tile (32x32), 224/32 = 7 exact
constexpr int HALO = 4;             // 2 each side
constexpr int LT = TS + HALO;       // 36 (row stride 144B; even -> b64-safe)
constexpr int NTX = Ww / TS;        // 7 tiles in x
constexpr int PLANE = Hh * Ww;      // 50176
constexpr int NTHREADS = 256;       // 8 waves (wave32)
constexpr int PAIRS_PER_ROW = LT / 2;            // 18
constexpr int NPAIRS = LT * PAIRS_PER_ROW;       // 648 b64 transfers per tile
}

// Types matching the async builtins' signatures (probe-confirmed):
//   b32: (AS1 int*, AS3 int*, imm, imm)
//   b64: (AS1 v2i*, AS3 v2i*, imm, imm)  where v2i = int vector_size(8)
typedef int v2i32 __attribute__((vector_size(8)));
typedef __attribute__((address_space(1))) int   gint_as1;
typedef __attribute__((address_space(3))) int   lint_as3;
typedef __attribute__((address_space(1))) v2i32 gv2_as1;
typedef __attribute__((address_space(3))) v2i32 lv2_as3;

// Async DMA 8 bytes global->LDS (per-lane).  LDS byte offset is the low 32
// bits of the flat shared-aperture address (ISA 10.2 aperture rules).
__device__ __forceinline__ void async_copy_b64(const float* gsrc, const float* ldst) {
  uint64_t ga = (uint64_t)(uintptr_t)gsrc;
  uint32_t la = (uint32_t)(uintptr_t)ldst;   // LDS byte offset
#if __has_builtin(__builtin_amdgcn_global_load_async_to_lds_b64)
  __builtin_amdgcn_global_load_async_to_lds_b64(
      (gv2_as1*)(uintptr_t)ga, (lv2_as3*)(uintptr_t)la,
      /*offset=*/0, /*cpol=*/0);
#elif __has_builtin(__builtin_amdgcn_global_load_async_to_lds_b32)
  __builtin_amdgcn_global_load_async_to_lds_b32(
      (gint_as1*)(uintptr_t)ga, (lint_as3*)(uintptr_t)la, 0, 0);
  __builtin_amdgcn_global_load_async_to_lds_b32(
      (gint_as1*)(uintptr_t)(ga + 4), (lint_as3*)(uintptr_t)(la + 4), 0, 0);
#else
  asm volatile("global_load_async_to_lds_b64 %0, %1, off"
               :: "v"(la), "v"(ga) : "memory");
#endif
}

__device__ __forceinline__ void wait_async_zero() {
#if __has_builtin(__builtin_amdgcn_s_wait_asynccnt)
  __builtin_amdgcn_s_wait_asynccnt(0);
#else
  asm volatile("s_wait_asynccnt 0" ::: "memory");
#endif
}

__global__ __launch_bounds__(NTHREADS)
void dwconv5x5_kernel(const float* __restrict__ x,
                      const float* __restrict__ kern,
                      const float* __restrict__ bias,
                      float* __restrict__ out) {
  __shared__ float tile[LT * LT];   // 5184 B

  const int tid   = threadIdx.x;
  const int plane = blockIdx.y;               // b*C + c
  const int c     = plane & (Cc - 1);
  const int ty    = blockIdx.x / NTX;
  const int tx    = blockIdx.x - ty * NTX;
  const int y0    = ty * TS, x0 = tx * TS;

  const float* xp = x + (size_t)plane * PLANE;

  // ---- per-channel weights + bias first: s_loads overlap the DMA below ----
  const float* kp = kern + c * 25;
  float w[25];
  #pragma unroll
  for (int i = 0; i < 25; ++i) w[i] = kp[i];
  const float bv = bias[c];

  // ---- stage 36x36 halo tile into LDS as 648 b64 async copies ----
  // x0 is a multiple of 32 and pairs start at even lx, so every 8B pair is
  // entirely in-bounds or entirely out-of-bounds in x (no partial pairs).
  #pragma unroll
  for (int it = 0; it < (NPAIRS + NTHREADS - 1) / NTHREADS; ++it) {
    int p = tid + it * NTHREADS;
    if (p < NPAIRS) {
      int ly = p / PAIRS_PER_ROW;
      int lx = (p - ly * PAIRS_PER_ROW) * 2;
      int gy = y0 - 2 + ly, gx = x0 - 2 + lx;
      int i  = ly * LT + lx;                       // even -> 8B-aligned LDS
      if ((unsigned)gy < (unsigned)Hh && (unsigned)gx < (unsigned)(Ww - 1)) {
        async_copy_b64(xp + gy * Ww + gx, &tile[i]);
      } else {
        *(float2*)&tile[i] = make_float2(0.0f, 0.0f);   // zero padding
      }
    }
  }
  wait_async_zero();       // ASYNCcnt: our DMA writes to LDS are done
  __syncthreads();         // + DScnt for the zero-fills, all waves visible

  // ---- compute: wave wv owns rows 4*wv..4*wv+3, lane = column ----
  const int lane = tid & 31;
  const int wv   = tid >> 5;
  const int ry   = wv * 4;

  float acc[4] = {bv, bv, bv, bv};
  #pragma unroll
  for (int kx = 0; kx < 5; ++kx) {
    float v[8];                                 // 8 input rows feed 4 outputs
    #pragma unroll
    for (int j = 0; j < 8; ++j)
      v[j] = tile[(ry + j) * LT + lane + kx];   // conflict-free banks
    #pragma unroll
    for (int r = 0; r < 4; ++r)
      #pragma unroll
      for (int ky = 0; ky < 5; ++ky)
        acc[r] = fmaf(w[ky * 5 + kx], v[r + ky], acc[r]);
  }

  // ---- coalesced stores: each wave writes 4 contiguous 128B rows ----
  float* op = out + (size_t)plane * PLANE + (size_t)(y0 + ry) * Ww + (x0 + lane);
  #pragma unroll
  for (int r = 0; r < 4; ++r) op[r * Ww] = acc[r];
}

extern "C" void kernel_launch(void* const* d_in, const int* in_sizes, int n_in,
                              void* d_out, int out_size, void* d_ws, size_t ws_size,
                              hipStream_t stream) {
  const float* x  = (const float*)d_in[0];   // (16,128,224,224)
  const float* k  = (const float*)d_in[1];   // (128,5,5)
  const float* b  = (const float*)d_in[2];   // (128,)
  float* out = (float*)d_out;                // (16,128,224,224)

  dim3 grid((Hh / TS) * NTX, Bn * Cc);       // (49, 2048)
  dwconv5x5_kernel<<<grid, NTHREADS, 0, stream>>>(x, k, b, out);
}